// CoxTimeLoss_82892868813301
// MI455X (gfx1250) — compile-verified
//
#include <hip/hip_runtime.h>
#include <math.h>

// ---------------------------------------------------------------------------
// CoxTimeLoss on MI455X (gfx1250)
//   B = 262144 samples, K = 128 bins. Memory-bound: one streaming pass over
//   the 128 MiB logits matrix (~5.6 us at 23.3 TB/s). Online (running-max)
//   logsumexp per column so logits are touched exactly once. Logits are
//   staged HBM -> LDS with the CDNA5 Tensor Data Mover (double-buffered,
//   TENSORcnt synchronized) so the VALU loop reads from LDS while the DMA
//   engine streams the next tile.
// ---------------------------------------------------------------------------

#define KBINS 128
#define ROWS_PER_BLOCK 512
#define ROWS_TILE 64
#define NTILES (ROWS_PER_BLOCK / ROWS_TILE)   // 8

#if defined(__has_builtin)
#if __has_builtin(__builtin_amdgcn_tensor_load_to_lds) && \
    __has_builtin(__builtin_amdgcn_s_wait_tensorcnt)
#define USE_TDM 1
#endif
#endif
#ifndef USE_TDM
#define USE_TDM 0
#endif

typedef unsigned int u32x4 __attribute__((ext_vector_type(4)));
typedef int          i32x8 __attribute__((ext_vector_type(8)));
typedef int          i32x4 __attribute__((ext_vector_type(4)));

__device__ __forceinline__ unsigned lds_byte_addr(const void* p) {
  // generic pointer to LDS: low 32 bits are the LDS byte offset
  return (unsigned)(size_t)p;
}

#if USE_TDM
// Issue one TDM load of a ROWS_TILE x KBINS f32 tile (row-major, stride KBINS)
// from global memory into LDS. Descriptor per CDNA5 ISA section 8 (D#).
// clang-23 toolchain: 6-arg builtin (g0 x4, g1 x8, g2 x4, g3 x4, extra x8, cpol).
__device__ __forceinline__ void tdm_load_tile(const float* gsrc, unsigned lds_addr) {
  unsigned long long ga = (unsigned long long)(size_t)gsrc;
  u32x4 g0;
  g0.x = 1u;                                   // count=1, user mode, no gather
  g0.y = lds_addr;                             // lds_addr[31:0]
  g0.z = (unsigned)(ga & 0xFFFFFFFFu);         // global_addr[31:0]
  g0.w = (unsigned)((ga >> 32) & 0x01FFFFFFu)  // global_addr[56:32]
       | (2u << 30);                           // type = 2 ("image")
  i32x8 g1;
  g1.s0 = (int)(2u << 16);                     // wg_mask=0, data_size=2 (4B)
  g1.s1 = (int)((unsigned)KBINS << 16);        // tensor_dim0[15:0] @ bits[63:48]
  g1.s2 = (int)((unsigned)ROWS_TILE << 16);    // tensor_dim1[15:0] @ bits[95:80]
  g1.s3 = (int)((unsigned)KBINS << 16);        // tile_dim0 @ bits[127:112]
  g1.s4 = (int)ROWS_TILE;                      // tile_dim1 @ bits[143:128], tile_dim2=0
  g1.s5 = (int)KBINS;                          // tensor_dim0_stride[31:0]
  g1.s6 = 0;                                   // stride0 hi / stride1 lo
  g1.s7 = 0;                                   // stride1 hi
  i32x4 gz4 = {0, 0, 0, 0};                    // 2D tensor: groups 2/3 unused
  i32x8 gz8 = {0, 0, 0, 0, 0, 0, 0, 0};        // trailing group (unused) zeroed
  __builtin_amdgcn_tensor_load_to_lds(g0, g1, gz4, gz4, gz8, /*cpol=*/0);
}
#endif

// ---------------------------------------------------------------------------
// Kernel 1: per-block partials. Thread t owns column t. Online masked
// logsumexp (m = running max, s = running sum of exp(x-m)) over this block's
// 512 rows, plus per-bin event count / numerator (the lane with t == label
// holds logits[i, label_i], so no atomics are needed).
// ---------------------------------------------------------------------------
__global__ __launch_bounds__(KBINS) void cox_partial_kernel(
    const float* __restrict__ logits, const int* __restrict__ labels,
    const int* __restrict__ events, float* __restrict__ pm,
    float* __restrict__ ps, float* __restrict__ pnev,
    float* __restrict__ pnum) {
  extern __shared__ char smem[];               // double-buffered logits tiles
  __shared__ int s_lab[2][ROWS_TILE];
  __shared__ int s_ev[2][ROWS_TILE];

  const int tid = threadIdx.x;                 // column index k
  const long long rowBase = (long long)blockIdx.x * ROWS_PER_BLOCK;

  float m = -INFINITY, s = 0.0f, nev = 0.0f, numer = 0.0f;

#if USE_TDM
  float* buf = (float*)smem;                   // [2][ROWS_TILE*KBINS]
  if (tid < 32) {                              // wave 0 drives the DMA engine
    tdm_load_tile(logits + rowBase * KBINS, lds_byte_addr(buf));
    tdm_load_tile(logits + (rowBase + ROWS_TILE) * KBINS,
                  lds_byte_addr(buf + ROWS_TILE * KBINS));
  }
#endif
  // labels/events for tile 0 (64 rows, loaded by threads 0..63)
  if (tid < ROWS_TILE) {
    s_lab[0][tid] = labels[rowBase + tid];
    s_ev[0][tid]  = events[rowBase + tid];
  }

  for (int t = 0; t < NTILES; ++t) {
#if USE_TDM
    if (tid < 32) {
      if (t + 1 < NTILES) __builtin_amdgcn_s_wait_tensorcnt((short)1);
      else                __builtin_amdgcn_s_wait_tensorcnt((short)0);
    }
#endif
    __syncthreads();                           // tile t data + labels visible

    if (t + 1 < NTILES && tid < ROWS_TILE) {   // prefetch next tile's labels
      const int nb = (t + 1) & 1;
      s_lab[nb][tid] = labels[rowBase + (long long)(t + 1) * ROWS_TILE + tid];
      s_ev[nb][tid]  = events[rowBase + (long long)(t + 1) * ROWS_TILE + tid];
    }

    const int cb = t & 1;
#if USE_TDM
    const float* tile = buf + cb * (ROWS_TILE * KBINS);
#else
    const float* tile = logits + (rowBase + (long long)t * ROWS_TILE) * KBINS;
#endif

#pragma unroll 4
    for (int r = 0; r < ROWS_TILE; ++r) {
      const int lab = s_lab[cb][r];            // LDS broadcast
      const float x = tile[r * KBINS + tid];   // coalesced / bank-conflict-free
      if (tid <= lab) {                        // risk set: labels_i >= k
        const float nm = fmaxf(m, x);
        s = s * __expf(m - nm) + __expf(x - nm);
        m = nm;
      }
      if (tid == lab) {                        // this lane holds own = x
        const float ev = (float)s_ev[cb][r];
        nev += ev;
        numer += ev * x;
      }
    }
    __syncthreads();                           // all waves done reading buf[cb]
#if USE_TDM
    if (t + 2 < NTILES && tid < 32) {
      tdm_load_tile(logits + (rowBase + (long long)(t + 2) * ROWS_TILE) * KBINS,
                    lds_byte_addr(buf + cb * (ROWS_TILE * KBINS)));
    }
#endif
  }

  const int o = blockIdx.x * KBINS + tid;
  pm[o] = m; ps[o] = s; pnev[o] = nev; pnum[o] = numer;
}

// ---------------------------------------------------------------------------
// Kernel 2: merge per-block partials (numerically-safe logsumexp combine),
// form the per-bin terms, reduce to the scalar loss.
// ---------------------------------------------------------------------------
__global__ __launch_bounds__(KBINS) void cox_finalize_kernel(
    const float* __restrict__ pm, const float* __restrict__ ps,
    const float* __restrict__ pnev, const float* __restrict__ pnum,
    int nblocks, float* __restrict__ out) {
  const int k = threadIdx.x;
  float M = -INFINITY, S = 0.0f, nev = 0.0f, numer = 0.0f;
  for (int b = 0; b < nblocks; ++b) {
    const int o = b * KBINS + k;
    const float mb = pm[o], sb = ps[o];
    if (sb > 0.0f) {                           // skip blocks with empty risk set
      const float nM = fmaxf(M, mb);
      S = S * __expf(M - nM) + sb * __expf(mb - nM);
      M = nM;
    }
    nev += pnev[o];
    numer += pnum[o];
  }
  const float term = (nev > 0.0f) ? (numer - nev * (M + logf(S))) : 0.0f;

  __shared__ float rt[KBINS];
  __shared__ float rn[KBINS];
  rt[k] = term; rn[k] = nev;
  __syncthreads();
#pragma unroll
  for (int off = KBINS / 2; off > 0; off >>= 1) {
    if (k < off) { rt[k] += rt[k + off]; rn[k] += rn[k + off]; }
    __syncthreads();
  }
  if (k == 0) out[0] = -rt[0] / fmaxf(rn[0], 1.0f);
}

// ---------------------------------------------------------------------------
extern "C" void kernel_launch(void* const* d_in, const int* in_sizes, int n_in,
                              void* d_out, int out_size, void* d_ws, size_t ws_size,
                              hipStream_t stream) {
  const float* logits = (const float*)d_in[0];
  const int* labels   = (const int*)d_in[1];
  const int* events   = (const int*)d_in[2];
  float* out          = (float*)d_out;

  const int B = in_sizes[1];                   // 262144
  const int nblocks = B / ROWS_PER_BLOCK;      // 512 (B divisible by 512)

  float* pm   = (float*)d_ws;                  // [nblocks][K] running max
  float* ps   = pm   + (size_t)nblocks * KBINS;
  float* pnev = ps   + (size_t)nblocks * KBINS;
  float* pnum = pnev + (size_t)nblocks * KBINS;  // total ws: 1 MiB

  const size_t shmem = (size_t)2 * ROWS_TILE * KBINS * sizeof(float); // 64 KiB

  cox_partial_kernel<<<nblocks, KBINS, shmem, stream>>>(
      logits, labels, events, pm, ps, pnev, pnum);
  cox_finalize_kernel<<<1, KBINS, 0, stream>>>(pm, ps, pnev, pnum, nblocks, out);
}